// Attention_16784732193182
// MI455X (gfx1250) — compile-verified
//
#include <hip/hip_runtime.h>
#include <stdint.h>

#define NUM_HEADS 32
#define NUM_KV    8
#define GQA       4
#define HEAD_DIM  128
#define BATCH     16
#define MAX_CTX   2048
#define WAVES     8
#define TILE      16
#define QK_SCALE  0.08838834764831845f   // 1/sqrt(128)

typedef float v2f __attribute__((ext_vector_type(2)));
typedef float v8f __attribute__((ext_vector_type(8)));

// Flash-decode paged attention, fp32 WMMA path (memory-bound kernel; fp32
// 16x16x4 WMMA keeps native precision and exactly matches the gather layout).
__global__ __launch_bounds__(256)
void paged_decode_kernel(const float* __restrict__ q,
                         const float* __restrict__ knew,
                         const float* __restrict__ vnew,
                         const float* __restrict__ kcache,
                         const float* __restrict__ vcache,
                         const long long* __restrict__ slot_map,
                         const long long* __restrict__ act_slots,
                         const long long* __restrict__ ctx_lens,
                         float* __restrict__ out)
{
    const int kvh  = blockIdx.x;          // 0..7
    const int b    = blockIdx.y;          // 0..15
    const int tid  = threadIdx.x;
    const int wave = tid >> 5;
    const int lane = tid & 31;
    const int ln   = lane & 15;           // column/row index within 16
    const int hi   = lane >> 4;           // half-wave select

    __shared__ float lds_m[WAVES][GQA];
    __shared__ float lds_s[WAVES][GQA];
    __shared__ float lds_o[WAVES][GQA][HEAD_DIM];

    const int len    = (int)ctx_lens[b];
    const int ntiles = (len + TILE - 1) / TILE;

    // slot_mapping values fit in int32 (NUM_SLOTS = 32784, or -1 = dropped)
    int sm32[BATCH];
#pragma unroll
    for (int j = 0; j < BATCH; ++j) sm32[j] = (int)slot_map[j];

    // Q in B-matrix layout: qb[c] = Q(head=ln, d=4c+2*hi, 4c+2*hi+1), 0 for pad heads
    v2f qb[32];
    {
        const bool live = (ln < GQA);
        const int  h    = live ? ln : 0;  // clamp so dead lanes read in-bounds
        const float* qrow = q + ((size_t)b * NUM_HEADS + (size_t)kvh * GQA + h) * HEAD_DIM;
#pragma unroll
        for (int c = 0; c < 32; ++c) {
            const int d = 4 * c + 2 * hi;
            v2f t = *(const v2f*)(qrow + d);
            qb[c][0] = live ? t[0] : 0.0f;
            qb[c][1] = live ? t[1] : 0.0f;
        }
    }

    // O^T accumulators: acc[dc][i] = O(d = 16*dc + i + 8*hi, head = ln)
    v8f acc[8];
#pragma unroll
    for (int dc = 0; dc < 8; ++dc)
#pragma unroll
        for (int i = 0; i < 8; ++i) acc[dc][i] = 0.0f;

    float m_run = -1e30f, s_run = 0.0f;

    for (int t = wave; t < ntiles; t += WAVES) {
        const int base = t * TILE;

        // this lane's slot row (ctx = base + ln), with new-token override
        const long long slot = act_slots[(size_t)b * MAX_CTX + base + ln];
        const int s32 = (int)slot;
        const float* krow = kcache + ((size_t)slot * NUM_KV + kvh) * HEAD_DIM;
        const float* vrow = vcache + ((size_t)slot * NUM_KV + kvh) * HEAD_DIM;
#pragma unroll
        for (int j = 0; j < BATCH; ++j) {
            if (s32 == sm32[j]) {
                krow = knew + ((size_t)j * NUM_KV + kvh) * HEAD_DIM;
                vrow = vnew + ((size_t)j * NUM_KV + kvh) * HEAD_DIM;
            }
        }

        // ---- S^T = K_tile(16x128) x Q^T(128x16heads), 32 x WMMA f32 16x16x4 ----
        v8f sc;
#pragma unroll
        for (int i = 0; i < 8; ++i) sc[i] = 0.0f;

#pragma unroll
        for (int half = 0; half < 2; ++half) {
            v2f kb[16];
#pragma unroll
            for (int c = 0; c < 16; ++c) {
                const int d = 4 * (half * 16 + c) + 2 * hi;   // 8B-aligned
                kb[c] = *(const v2f*)(krow + d);              // global_load_b64
            }
#pragma unroll
            for (int c = 0; c < 16; ++c) {
                sc = __builtin_amdgcn_wmma_f32_16x16x4_f32(
                        false, kb[c], false, qb[half * 16 + c],
                        (short)0, sc, false, false);
            }
        }

        // ---- online softmax: lane = head, vgprs = ctx rows (i + 8*hi) ----
        float p[8];
        float tmax = -1e30f;
#pragma unroll
        for (int i = 0; i < 8; ++i) {
            const int pos = base + i + 8 * hi;
            const float s = (pos < len) ? sc[i] * QK_SCALE : -1e30f;
            p[i] = s;
            tmax = fmaxf(tmax, s);
        }
        tmax = fmaxf(tmax, __shfl_xor(tmax, 16, 32));
        const float m_new = fmaxf(m_run, tmax);
        const float corr  = __expf(m_run - m_new);
        float tsum = 0.0f;
#pragma unroll
        for (int i = 0; i < 8; ++i) {
            p[i] = __expf(p[i] - m_new);
            tsum += p[i];
        }
        tsum += __shfl_xor(tsum, 16, 32);
        s_run = s_run * corr + tsum;
        m_run = m_new;
#pragma unroll
        for (int dc = 0; dc < 8; ++dc)
#pragma unroll
            for (int i = 0; i < 8; ++i) acc[dc][i] *= corr;

        // ---- O^T += V^T(16d x 4ctx) x P(4ctx x 16heads), 32 x WMMA ----
        const unsigned long long vp = (unsigned long long)(uintptr_t)vrow;
#pragma unroll
        for (int c = 0; c < 4; ++c) {
            // B = P chunk: b0 -> ctx 4c+2hi, b1 -> ctx 4c+1+2hi (half-swap shuffles)
            v2f pb;
            if (c < 2) {
                const float x0 = __shfl_xor(p[4 * c + 2], 16, 32);
                const float x1 = __shfl_xor(p[4 * c + 3], 16, 32);
                pb[0] = hi ? x0 : p[4 * c + 0];
                pb[1] = hi ? x1 : p[4 * c + 1];
            } else {
                const float x0 = __shfl_xor(p[4 * c - 8 + 0], 16, 32);
                const float x1 = __shfl_xor(p[4 * c - 8 + 1], 16, 32);
                pb[0] = hi ? p[4 * c - 8 + 2] : x0;
                pb[1] = hi ? p[4 * c - 8 + 3] : x1;
            }
            // V row pointers for ctx 4c+2hi and 4c+1+2hi (held by lane == ctx)
            const float* r0 = (const float*)(uintptr_t)__shfl(vp, 4 * c + 2 * hi, 32);
            const float* r1 = (const float*)(uintptr_t)__shfl(vp, 4 * c + 1 + 2 * hi, 32);
#pragma unroll
            for (int dc = 0; dc < 8; ++dc) {
                v2f va;
                va[0] = r0[dc * 16 + ln];
                va[1] = r1[dc * 16 + ln];
                acc[dc] = __builtin_amdgcn_wmma_f32_16x16x4_f32(
                              false, va, false, pb, (short)0, acc[dc], false, false);
            }
        }
    }

    // ---- per-wave partials -> LDS ----
    if (ln < GQA) {
        if (hi == 0) { lds_m[wave][ln] = m_run; lds_s[wave][ln] = s_run; }
#pragma unroll
        for (int dc = 0; dc < 8; ++dc)
#pragma unroll
            for (int i = 0; i < 8; ++i)
                lds_o[wave][ln][dc * 16 + i + 8 * hi] = acc[dc][i];
    }
    __syncthreads();

    // ---- combine 8 wave partials (flash-decode stage 2) ----
    if (tid < GQA * 32) {
        const int g = tid >> 5;
        const int l = tid & 31;
        float mg = -1e30f;
#pragma unroll
        for (int w = 0; w < WAVES; ++w) mg = fmaxf(mg, lds_m[w][g]);
        float cw[WAVES];
        float sg = 0.0f;
#pragma unroll
        for (int w = 0; w < WAVES; ++w) {
            cw[w] = __expf(lds_m[w][g] - mg);
            sg += cw[w] * lds_s[w][g];
        }
        const float inv = 1.0f / sg;
#pragma unroll
        for (int j = 0; j < 4; ++j) {
            const int d = l + 32 * j;
            float o = 0.0f;
#pragma unroll
            for (int w = 0; w < WAVES; ++w) o += cw[w] * lds_o[w][g][d];
            out[((size_t)b * NUM_HEADS + (size_t)kvh * GQA + g) * HEAD_DIM + d] = o * inv;
        }
    }
}

extern "C" void kernel_launch(void* const* d_in, const int* in_sizes, int n_in,
                              void* d_out, int out_size, void* d_ws, size_t ws_size,
                              hipStream_t stream) {
    const float*     q  = (const float*)d_in[0];
    const float*     k  = (const float*)d_in[1];
    const float*     v  = (const float*)d_in[2];
    const float*     kc = (const float*)d_in[3];
    const float*     vc = (const float*)d_in[4];
    const long long* sm = (const long long*)d_in[5];
    const long long* as = (const long long*)d_in[6];
    const long long* cl = (const long long*)d_in[7];
    float* o = (float*)d_out;

    dim3 grid(NUM_KV, BATCH);   // one WG per (kv_head, batch)
    paged_decode_kernel<<<grid, 256, 0, stream>>>(q, k, v, kc, vc, sm, as, cl, o);
}